// CrossAttentionGenerator_56831007261027
// MI455X (gfx1250) — compile-verified
//
#include <hip/hip_runtime.h>
#include <hip/hip_bf16.h>
#include <math.h>

typedef __attribute__((ext_vector_type(2))) float v2f;
typedef __attribute__((ext_vector_type(8))) float v8f;

#define F_DIM 64
#define LN_EPS 1e-5f
#define K_NN 8
#define CHUNK 2048

// ---------------------------------------------------------------------------
// Kernel 1: point projection  Linear(3->64) -> LayerNorm -> ReLU -> Linear(64->64)
// One wave handles 16 points. Hidden vector is built directly in the
// V_WMMA_F32_16X16X4_F32 A-fragment layout:
//   lane L (0..31): point = L%16, half = L/16
//   a[c] (c=0..15) holds hidden channels k = 4c + 2*half + {0,1}
// Second linear done with 16 K-steps x 4 N-tiles of wmma_f32_16x16x4_f32.
// ---------------------------------------------------------------------------
__global__ __launch_bounds__(256) void proj_kernel(
    const float* __restrict__ pts,   // (P,3)
    const float* __restrict__ W1,    // (3,64)
    const float* __restrict__ b1,    // (64)
    const float* __restrict__ ln_g,  // (64)
    const float* __restrict__ ln_b,  // (64)
    const float* __restrict__ W2,    // (64,64)
    const float* __restrict__ b2,    // (64)
    float* __restrict__ feat)        // (P,64)
{
    __shared__ float sW2[F_DIM * F_DIM];
    __shared__ float sW1[3 * F_DIM];
    __shared__ float sb1[F_DIM], sg[F_DIM], sbt[F_DIM], sb2[F_DIM];

    const int tid = threadIdx.x;
    for (int i = tid; i < F_DIM * F_DIM; i += 256) sW2[i] = W2[i];
    for (int i = tid; i < 3 * F_DIM; i += 256)     sW1[i] = W1[i];
    if (tid < F_DIM) {
        sb1[tid] = b1[tid];
        sg[tid]  = ln_g[tid];
        sbt[tid] = ln_b[tid];
        sb2[tid] = b2[tid];
    }
    __syncthreads();

    const int wave = tid >> 5;        // 0..7
    const int lane = tid & 31;
    const int half = lane >> 4;       // 0 or 1 (K half / M half)
    const int pl   = lane & 15;       // 0..15
    const int tile_base = (blockIdx.x * 8 + wave) * 16;
    const int P = tile_base + pl;     // point for this lane (both halves same)

    const float x = pts[P * 3 + 0];
    const float y = pts[P * 3 + 1];
    const float z = pts[P * 3 + 2];

    // ---- Linear(3->64) into A-fragment layout ----
    v2f a[16];
    float sum = 0.f;
#pragma unroll
    for (int c = 0; c < 16; ++c) {
        const int k0 = 4 * c + 2 * half;
        float h0 = x * sW1[k0]     + y * sW1[64 + k0]     + z * sW1[128 + k0]     + sb1[k0];
        float h1 = x * sW1[k0 + 1] + y * sW1[64 + k0 + 1] + z * sW1[128 + k0 + 1] + sb1[k0 + 1];
        a[c].x = h0; a[c].y = h1;
        sum += h0 + h1;
    }

    // ---- LayerNorm over 64 channels (this lane holds 32, partner lane^16 the rest) ----
    sum += __shfl_xor(sum, 16);
    const float mu = sum * (1.0f / 64.0f);
    float s2 = 0.f;
#pragma unroll
    for (int c = 0; c < 16; ++c) {
        float dx = a[c].x - mu; s2 += dx * dx;
        float dy = a[c].y - mu; s2 += dy * dy;
    }
    s2 += __shfl_xor(s2, 16);
    const float rstd = rsqrtf(s2 * (1.0f / 64.0f) + LN_EPS);

#pragma unroll
    for (int c = 0; c < 16; ++c) {
        const int k0 = 4 * c + 2 * half;
        float n0 = (a[c].x - mu) * rstd * sg[k0]     + sbt[k0];
        float n1 = (a[c].y - mu) * rstd * sg[k0 + 1] + sbt[k0 + 1];
        a[c].x = fmaxf(n0, 0.0f);   // ReLU
        a[c].y = fmaxf(n1, 0.0f);
    }

    // ---- Linear(64->64) via WMMA f32 16x16x4 ----
    v8f acc0 = {}, acc1 = {}, acc2 = {}, acc3 = {};
#pragma unroll
    for (int c = 0; c < 16; ++c) {
        const int krow = 4 * c + 2 * half;   // rows of W2 held by this lane
#pragma unroll
        for (int t = 0; t < 4; ++t) {
            v2f bb;
            const int ncol = t * 16 + pl;
            bb.x = sW2[krow * 64 + ncol];
            bb.y = sW2[(krow + 1) * 64 + ncol];
            v8f* accp = (t == 0) ? &acc0 : (t == 1) ? &acc1 : (t == 2) ? &acc2 : &acc3;
            *accp = __builtin_amdgcn_wmma_f32_16x16x4_f32(
                /*neg_a=*/false, a[c], /*neg_b=*/false, bb,
                /*c_mod=*/(short)0, *accp, /*reuse_a=*/false, /*reuse_b=*/false);
        }
    }

    // ---- bias + store: D vgpr r holds row (r + 8*half), col t*16 + pl ----
#pragma unroll
    for (int t = 0; t < 4; ++t) {
        const int n = t * 16 + pl;
        const float bias = sb2[n];
        const v8f accv = (t == 0) ? acc0 : (t == 1) ? acc1 : (t == 2) ? acc2 : acc3;
#pragma unroll
        for (int r = 0; r < 8; ++r) {
            const int m = tile_base + r + 8 * half;
            feat[m * 64 + n] = accv[r] + bias;
        }
    }
}

// ---------------------------------------------------------------------------
// Kernel 2: fused kNN (top-8 smallest squared distances) + softmax attention.
// One thread per source point; target positions (+|t|^2) staged through LDS.
// ---------------------------------------------------------------------------
__global__ __launch_bounds__(256) void knn_attn_kernel(
    const float* __restrict__ src,    // (B*N,3)
    const float* __restrict__ tgt,    // (B*M,3)
    const float* __restrict__ sfeat,  // (B*N,64)
    const float* __restrict__ tfeat,  // (B*M,64)
    const float* __restrict__ log_temp,
    float* __restrict__ out,          // (B*N,3)
    int Npts, int Mpts)
{
    __shared__ float4 sT[CHUNK];      // x, y, z, |t|^2   (32 KB)

    const int tid = threadIdx.x;
    const int gp  = blockIdx.x * 256 + tid;       // global source point
    const int b   = gp / Npts;                    // batch (block-uniform: N % 256 == 0)

    const float sx = src[gp * 3 + 0];
    const float sy = src[gp * 3 + 1];
    const float sz = src[gp * 3 + 2];
    const float ss = sx * sx + sy * sy + sz * sz;

    const float* tb = tgt + (long)b * Mpts * 3;

    float bd[K_NN];
    int   bi[K_NN];
#pragma unroll
    for (int k = 0; k < K_NN; ++k) { bd[k] = 3.402823466e38f; bi[k] = 0; }

    for (int c0 = 0; c0 < Mpts; c0 += CHUNK) {
        __syncthreads();
        for (int i = tid; i < CHUNK; i += 256) {
            const float tx = tb[(c0 + i) * 3 + 0];
            const float ty = tb[(c0 + i) * 3 + 1];
            const float tz = tb[(c0 + i) * 3 + 2];
            sT[i] = make_float4(tx, ty, tz, tx * tx + ty * ty + tz * tz);
        }
        __syncthreads();

        for (int i = 0; i < CHUNK; ++i) {
            const float4 t = sT[i];
            const float cross = sx * t.x + sy * t.y + sz * t.z;
            const float d = ss + t.w - 2.0f * cross;
            if (d < bd[K_NN - 1]) {             // strict: ties keep lower index
                bd[K_NN - 1] = d;
                bi[K_NN - 1] = c0 + i;
#pragma unroll
                for (int q = K_NN - 1; q > 0; --q) {
                    if (bd[q] < bd[q - 1]) {    // strict: stable among equals
                        float td = bd[q]; bd[q] = bd[q - 1]; bd[q - 1] = td;
                        int   ti = bi[q]; bi[q] = bi[q - 1]; bi[q - 1] = ti;
                    }
                }
            }
        }
    }

    // ---- attention over the 8 neighbors ----
    const float temperature = expf(log_temp[0]) * 8.0f;   // sqrt(64)
    const float4* sf4 = (const float4*)(sfeat + (long)gp * 64);
    float4 sfr[16];
#pragma unroll
    for (int f = 0; f < 16; ++f) sfr[f] = sf4[f];

    float logit[K_NN];
#pragma unroll
    for (int k = 0; k < K_NN; ++k) {
        const float4* tf4 = (const float4*)(tfeat + ((long)b * Mpts + bi[k]) * 64);
        float acc = 0.f;
#pragma unroll
        for (int f = 0; f < 16; ++f) {
            const float4 t4 = tf4[f];
            acc += sfr[f].x * t4.x + sfr[f].y * t4.y + sfr[f].z * t4.z + sfr[f].w * t4.w;
        }
        logit[k] = acc / temperature;
    }

    float mx = logit[0];
#pragma unroll
    for (int k = 1; k < K_NN; ++k) mx = fmaxf(mx, logit[k]);
    float w[K_NN];
    float se = 0.f;
#pragma unroll
    for (int k = 0; k < K_NN; ++k) { w[k] = expf(logit[k] - mx); se += w[k]; }
    const float inv = 1.0f / se;

    float ax = 0.f, ay = 0.f, az = 0.f;
#pragma unroll
    for (int k = 0; k < K_NN; ++k) {
        const float wk = w[k] * inv;
        const float* tp = tb + bi[k] * 3;
        ax += wk * tp[0];
        ay += wk * tp[1];
        az += wk * tp[2];
    }
    out[gp * 3 + 0] = ax;
    out[gp * 3 + 1] = ay;
    out[gp * 3 + 2] = az;
}

// ---------------------------------------------------------------------------
extern "C" void kernel_launch(void* const* d_in, const int* in_sizes, int n_in,
                              void* d_out, int out_size, void* d_ws, size_t ws_size,
                              hipStream_t stream) {
    const float* source   = (const float*)d_in[0];
    const float* target   = (const float*)d_in[1];
    const float* W1       = (const float*)d_in[2];
    const float* b1       = (const float*)d_in[3];
    const float* ln_g     = (const float*)d_in[4];
    const float* ln_b     = (const float*)d_in[5];
    const float* W2       = (const float*)d_in[6];
    const float* b2       = (const float*)d_in[7];
    const float* log_temp = (const float*)d_in[8];
    float* out = (float*)d_out;

    const int B = 4;
    const int Psrc = in_sizes[0] / 3;   // B*N = 32768
    const int Ptgt = in_sizes[1] / 3;   // B*M = 32768
    const int N = Psrc / B;
    const int M = Ptgt / B;

    float* sfeat = (float*)d_ws;
    float* tfeat = sfeat + (size_t)Psrc * F_DIM;

    // 128 points per block (8 waves x 16 points)
    proj_kernel<<<Psrc / 128, 256, 0, stream>>>(source, W1, b1, ln_g, ln_b, W2, b2, sfeat);
    proj_kernel<<<Ptgt / 128, 256, 0, stream>>>(target, W1, b1, ln_g, ln_b, W2, b2, tfeat);

    knn_attn_kernel<<<Psrc / 256, 256, 0, stream>>>(source, target, sfeat, tfeat,
                                                    log_temp, out, N, M);
}